// DistNSA_8366596292685
// MI455X (gfx1250) — compile-verified
//
#include <hip/hip_runtime.h>
#include <hip/hip_bf16.h>

#define SS   2048
#define NHQ  8
#define NHK  2
#define HDIM 128
#define REPG 4
#define WINW 512
#define BLKC 32
#define NBC  64
#define TOPN 4
#define NEGV (-1e30f)

typedef __attribute__((ext_vector_type(2))) float v2f;
typedef __attribute__((ext_vector_type(8))) float v8f;

#if defined(__has_builtin)
#if __has_builtin(__builtin_amdgcn_global_load_async_to_lds_b128)
#define HAVE_ASYNC_LDS 1
#endif
#endif

#ifdef HAVE_ASYNC_LDS
// builtin signature (from hipcc diagnostic): takes pointers to 16-byte int
// vectors: (v4i addrspace(1)*, v4i addrspace(3)*, imm offset, imm cpol)
typedef int v4i_ __attribute__((vector_size(16)));
typedef __attribute__((address_space(1))) v4i_ gv4i;
typedef __attribute__((address_space(3))) v4i_ lv4i;
__device__ __forceinline__ void async_wait_all() {
#if __has_builtin(__builtin_amdgcn_s_wait_asynccnt)
    __builtin_amdgcn_s_wait_asynccnt(0);
#else
    asm volatile("s_wait_asynccnt 0" ::: "memory");
#endif
}
#endif

// --------------------------------------------------------------------------
// Kernel 1: block-mean compressed K/V.  kc/vc: [NB][NHK][HD]
// --------------------------------------------------------------------------
__global__ void nsa_cmpkv_kernel(const float* __restrict__ k,
                                 const float* __restrict__ v,
                                 float* __restrict__ kc,
                                 float* __restrict__ vc) {
    int idx = blockIdx.x * blockDim.x + threadIdx.x;
    if (idx >= NBC * NHK * HDIM) return;
    int d  = idx % HDIM;
    int hk = (idx / HDIM) % NHK;
    int n  = idx / (HDIM * NHK);
    float sk = 0.f, sv = 0.f;
    for (int i = 0; i < BLKC; ++i) {
        int row = n * BLKC + i;
        sk += k[(row * NHK + hk) * HDIM + d];
        sv += v[(row * NHK + hk) * HDIM + d];
    }
    kc[(n * NHK + hk) * HDIM + d] = sk * (1.0f / BLKC);
    vc[(n * NHK + hk) * HDIM + d] = sv * (1.0f / BLKC);
}

// --------------------------------------------------------------------------
// Kernel 2: sliding-window branch, flash attention, WMMA f32 16x16x4.
// One wave per (16-query tile, q-head). Double-buffered async global->LDS
// staging of K/V tiles overlaps the next tile's load with this tile's WMMAs.
// Writes (initializes) d_out.
// --------------------------------------------------------------------------
__global__ __launch_bounds__(32)
void nsa_win_kernel(const float* __restrict__ q, const float* __restrict__ k,
                    const float* __restrict__ v, const float* __restrict__ gw,
                    float* __restrict__ out) {
    __shared__ __align__(16) float q_l[16][132];
    __shared__ __align__(16) float k_l[2][16][132];
    __shared__ __align__(16) float v_l[2][16][132];
    __shared__ float p_l[16][17];

    const int qt   = blockIdx.x / NHQ;
    const int hq   = blockIdx.x % NHQ;
    const int hk   = hq / REPG;
    const int q0   = qt * 16;
    const int tid  = threadIdx.x;
    const int half = tid >> 4;      // 0 or 1
    const int l16  = tid & 15;
    const float scale = 0.08838834764831845f;   // 1/sqrt(128)

    const int ktd = qt;
    int lowkey = q0 - (WINW - 1); if (lowkey < 0) lowkey = 0;
    const int ktl = lowkey >> 4;

    // ---- tile staging (async double-buffer if available) ----
#ifdef HAVE_ASYNC_LDS
#define ISSUE_TILE(bufi, ktv)                                                  \
    do {                                                                       \
        for (int i_ = 0; i_ < 16; ++i_) {                                      \
            int idx_ = i_ * 32 + tid;          /* 512 16B chunks per tile */   \
            int r_ = idx_ >> 5;                                                \
            int d_ = (idx_ & 31) << 2;                                         \
            const float* gk_ = &k[(((ktv) * 16 + r_) * NHK + hk) * HDIM + d_]; \
            const float* gv_ = &v[(((ktv) * 16 + r_) * NHK + hk) * HDIM + d_]; \
            __builtin_amdgcn_global_load_async_to_lds_b128(                    \
                (gv4i*)gk_, (lv4i*)&k_l[bufi][r_][d_], 0, 0);                  \
            __builtin_amdgcn_global_load_async_to_lds_b128(                    \
                (gv4i*)gv_, (lv4i*)&v_l[bufi][r_][d_], 0, 0);                  \
        }                                                                      \
    } while (0)
#define WAIT_TILE() async_wait_all()
#else
#define ISSUE_TILE(bufi, ktv)                                                  \
    do {                                                                       \
        for (int idx_ = tid; idx_ < 16 * HDIM; idx_ += 32) {                   \
            int r_ = idx_ >> 7, d_ = idx_ & 127;                               \
            k_l[bufi][r_][d_] = k[(((ktv) * 16 + r_) * NHK + hk) * HDIM + d_]; \
            v_l[bufi][r_][d_] = v[(((ktv) * 16 + r_) * NHK + hk) * HDIM + d_]; \
        }                                                                      \
    } while (0)
#define WAIT_TILE() do { } while (0)
#endif

    int buf = 0;
    ISSUE_TILE(0, ktd);   // prologue: first (diagonal) tile in flight

    for (int idx = tid; idx < 16 * HDIM; idx += 32) {
        int r = idx >> 7, d = idx & 127;
        q_l[r][d] = q[((q0 + r) * NHQ + hq) * HDIM + d];
    }
    __syncthreads();

    v8f acc[8];
    float mrow[8], lrow[8];
#pragma unroll
    for (int nt = 0; nt < 8; ++nt)
#pragma unroll
        for (int j = 0; j < 8; ++j) acc[nt][j] = 0.f;
#pragma unroll
    for (int j = 0; j < 8; ++j) { mrow[j] = -3.0e38f; lrow[j] = 0.f; }

    for (int kt = ktd; kt >= ktl; --kt) {
        WAIT_TILE();            // buffer `buf` now resident in LDS
        __syncthreads();
        const int nxt = buf ^ 1;
        if (kt - 1 >= ktl) ISSUE_TILE(nxt, kt - 1);  // overlap with compute

        // ---- S = Q K^T (16x16 tile) ----
        v8f c;
#pragma unroll
        for (int j = 0; j < 8; ++j) c[j] = 0.f;
#pragma unroll
        for (int kk = 0; kk < HDIM; kk += 4) {
            v2f a, b;
            a[0] = q_l[l16][kk + 2 * half + 0];
            a[1] = q_l[l16][kk + 2 * half + 1];
            b[0] = k_l[buf][l16][kk + 2 * half + 0];
            b[1] = k_l[buf][l16][kk + 2 * half + 1];
            c = __builtin_amdgcn_wmma_f32_16x16x4_f32(false, a, false, b,
                                                      (short)0, c, false, false);
        }

        // ---- mask + online softmax update ----
        const int ng = kt * 16 + l16;
#pragma unroll
        for (int j = 0; j < 8; ++j) {
            int qs  = q0 + j + 8 * half;
            int dif = qs - ng;
            bool valid = (dif >= 0) && (dif < WINW);
            float sv = valid ? c[j] * scale : NEGV;
            float rmax = sv;
#pragma unroll
            for (int m = 1; m <= 8; m <<= 1)
                rmax = fmaxf(rmax, __shfl_xor(rmax, m, 32));
            float mnew = fmaxf(mrow[j], rmax);
            float corr = expf(mrow[j] - mnew);
            float p    = expf(sv - mnew);
            float rsum = p;
#pragma unroll
            for (int m = 1; m <= 8; m <<= 1)
                rsum += __shfl_xor(rsum, m, 32);
            lrow[j] = lrow[j] * corr + rsum;
            mrow[j] = mnew;
#pragma unroll
            for (int nt = 0; nt < 8; ++nt) acc[nt][j] *= corr;
            p_l[j + 8 * half][l16] = p;
        }
        __syncthreads();

        // ---- O += P * V (16x16 x 16x128) ----
#pragma unroll
        for (int kc4 = 0; kc4 < 16; kc4 += 4) {
            v2f a;
            a[0] = p_l[l16][kc4 + 2 * half + 0];
            a[1] = p_l[l16][kc4 + 2 * half + 1];
#pragma unroll
            for (int nt = 0; nt < 8; ++nt) {
                v2f b;
                b[0] = v_l[buf][kc4 + 2 * half + 0][nt * 16 + l16];
                b[1] = v_l[buf][kc4 + 2 * half + 1][nt * 16 + l16];
                acc[nt] = __builtin_amdgcn_wmma_f32_16x16x4_f32(false, a, false, b,
                                                   (short)0, acc[nt], false, false);
            }
        }
        __syncthreads();
        buf = nxt;
    }

    for (int nt = 0; nt < 8; ++nt)
        for (int j = 0; j < 8; ++j) {
            int row = q0 + j + 8 * half;
            int d   = nt * 16 + l16;
            float o = acc[nt][j] / lrow[j];
            out[(row * NHQ + hq) * HDIM + d] = gw[row * NHQ + hq] * o;
        }
#undef ISSUE_TILE
#undef WAIT_TILE
}

// --------------------------------------------------------------------------
// Kernel 3: compressed branch. Writes normalized pc [NHQ][S][NB] to ws and
// adds g_cmp * o_cmp into d_out.
// --------------------------------------------------------------------------
__global__ __launch_bounds__(32)
void nsa_cmp_kernel(const float* __restrict__ q, const float* __restrict__ kc,
                    const float* __restrict__ vc, const float* __restrict__ gc,
                    float* __restrict__ pc_ws, float* __restrict__ out) {
    __shared__ float q_l[16][132];
    __shared__ float k_l[16][132];
    __shared__ float v_l[16][132];
    __shared__ float p_l[16][68];

    const int qt   = blockIdx.x / NHQ;
    const int hq   = blockIdx.x % NHQ;
    const int hk   = hq / REPG;
    const int q0   = qt * 16;
    const int tid  = threadIdx.x;
    const int half = tid >> 4;
    const int l16  = tid & 15;
    const float scale = 0.08838834764831845f;

    for (int idx = tid; idx < 16 * HDIM; idx += 32) {
        int r = idx >> 7, d = idx & 127;
        q_l[r][d] = q[((q0 + r) * NHQ + hq) * HDIM + d];
    }

    v8f sc[4];
    for (int nt = 0; nt < 4; ++nt) {
        __syncthreads();
        for (int idx = tid; idx < 16 * HDIM; idx += 32) {
            int r = idx >> 7, d = idx & 127;
            k_l[r][d] = kc[((nt * 16 + r) * NHK + hk) * HDIM + d];
        }
        __syncthreads();
        v8f c;
#pragma unroll
        for (int j = 0; j < 8; ++j) c[j] = 0.f;
#pragma unroll
        for (int kk = 0; kk < HDIM; kk += 4) {
            v2f a, b;
            a[0] = q_l[l16][kk + 2 * half + 0];
            a[1] = q_l[l16][kk + 2 * half + 1];
            b[0] = k_l[l16][kk + 2 * half + 0];
            b[1] = k_l[l16][kk + 2 * half + 1];
            c = __builtin_amdgcn_wmma_f32_16x16x4_f32(false, a, false, b,
                                                      (short)0, c, false, false);
        }
        sc[nt] = c;
    }

    // mask + scale  (block fully in the past: (n+1)*32-1 <= s)
    for (int nt = 0; nt < 4; ++nt)
#pragma unroll
        for (int j = 0; j < 8; ++j) {
            int n = nt * 16 + l16;
            int s = q0 + j + 8 * half;
            bool valid = ((n + 1) * BLKC - 1) <= s;
            sc[nt][j] = valid ? sc[nt][j] * scale : NEGV;
        }

    // row softmax over 64 blocks
    float mrow[8], lrow[8];
#pragma unroll
    for (int j = 0; j < 8; ++j) {
        float rm = -3.0e38f;
        for (int nt = 0; nt < 4; ++nt) rm = fmaxf(rm, sc[nt][j]);
#pragma unroll
        for (int m = 1; m <= 8; m <<= 1) rm = fmaxf(rm, __shfl_xor(rm, m, 32));
        mrow[j] = rm;
    }
    for (int nt = 0; nt < 4; ++nt)
#pragma unroll
        for (int j = 0; j < 8; ++j) sc[nt][j] = expf(sc[nt][j] - mrow[j]);
#pragma unroll
    for (int j = 0; j < 8; ++j) {
        float rs = 0.f;
        for (int nt = 0; nt < 4; ++nt) rs += sc[nt][j];
#pragma unroll
        for (int m = 1; m <= 8; m <<= 1) rs += __shfl_xor(rs, m, 32);
        lrow[j] = rs;
    }
    for (int nt = 0; nt < 4; ++nt)
#pragma unroll
        for (int j = 0; j < 8; ++j) {
            float p = sc[nt][j] / lrow[j];
            int s = q0 + j + 8 * half;
            int n = nt * 16 + l16;
            pc_ws[(hq * SS + s) * NBC + n] = p;
            p_l[j + 8 * half][n] = p;
        }

    // O = P (16x64) * vc (64x128), staged 16 K-rows at a time
    v8f acc[8];
#pragma unroll
    for (int nt = 0; nt < 8; ++nt)
#pragma unroll
        for (int j = 0; j < 8; ++j) acc[nt][j] = 0.f;

    for (int st = 0; st < 4; ++st) {
        __syncthreads();
        for (int idx = tid; idx < 16 * HDIM; idx += 32) {
            int r = idx >> 7, d = idx & 127;
            v_l[r][d] = vc[((st * 16 + r) * NHK + hk) * HDIM + d];
        }
        __syncthreads();
#pragma unroll
        for (int kq = 0; kq < 16; kq += 4) {
            int kc4 = st * 16 + kq;
            v2f a;
            a[0] = p_l[l16][kc4 + 2 * half + 0];
            a[1] = p_l[l16][kc4 + 2 * half + 1];
#pragma unroll
            for (int nt = 0; nt < 8; ++nt) {
                v2f b;
                b[0] = v_l[kq + 2 * half + 0][nt * 16 + l16];
                b[1] = v_l[kq + 2 * half + 1][nt * 16 + l16];
                acc[nt] = __builtin_amdgcn_wmma_f32_16x16x4_f32(false, a, false, b,
                                                   (short)0, acc[nt], false, false);
            }
        }
    }

    for (int nt = 0; nt < 8; ++nt)
        for (int j = 0; j < 8; ++j) {
            int row = q0 + j + 8 * half;
            int d   = nt * 16 + l16;
            float o = (row >= BLKC - 1) ? acc[nt][j] : 0.f;  // cmask.any zeroing
            out[(row * NHQ + hq) * HDIM + d] += gc[row * NHQ + hq] * o;
        }
}

// --------------------------------------------------------------------------
// Kernel 4: top-4 block selection per (s, hk). Deterministic argmax passes,
// strict '>' matches top_k's lower-index tie break.
// --------------------------------------------------------------------------
__global__ void nsa_sel_kernel(const float* __restrict__ pc, int* __restrict__ sel) {
    int idx = blockIdx.x * blockDim.x + threadIdx.x;
    if (idx >= SS * NHK) return;
    int hk = idx % NHK;
    int s  = idx / NHK;
    int cur = s / BLKC;
    float scb[NBC];
    for (int n = 0; n < NBC; ++n) {
        float vsum = 0.f;
        for (int r = 0; r < REPG; ++r)
            vsum += pc[((hk * REPG + r) * SS + s) * NBC + n];
        if (n == cur) vsum += 1e6f;
        if (n == 0)   vsum += 1e6f;
        scb[n] = vsum;
    }
    for (int t = 0; t < TOPN; ++t) {
        int best = 0; float bv = scb[0];
        for (int n = 1; n < NBC; ++n)
            if (scb[n] > bv) { bv = scb[n]; best = n; }
        sel[(s * NHK + hk) * TOPN + t] = best;
        scb[best] = -3.0e38f;
    }
}

// --------------------------------------------------------------------------
// Kernel 5: selected branch. 128 threads per (s, hk): 4 rep heads, 128
// gathered keys. Adds g_slt * o_slt into d_out.
// --------------------------------------------------------------------------
__global__ __launch_bounds__(128)
void nsa_slt_kernel(const float* __restrict__ q, const float* __restrict__ k,
                    const float* __restrict__ v, const float* __restrict__ gs,
                    const int* __restrict__ sel, float* __restrict__ out) {
    __shared__ float qsm[REPG][HDIM];
    __shared__ float scm[REPG][TOPN * BLKC];
    __shared__ int   selb[TOPN];

    const int s  = blockIdx.x / NHK;
    const int hk = blockIdx.x % NHK;
    const int t  = threadIdx.x;
    const float scale = 0.08838834764831845f;

    for (int i = t; i < REPG * HDIM; i += 128) {
        int r = i / HDIM, d = i % HDIM;
        qsm[r][d] = q[(s * NHQ + hk * REPG + r) * HDIM + d];
    }
    if (t < TOPN) selb[t] = sel[(s * NHK + hk) * TOPN + t];
    __syncthreads();

    // scores: thread t handles key t
    {
        int b = selb[t / BLKC];
        int pos = b * BLKC + (t % BLKC);
        const float* kr = k + (pos * NHK + hk) * HDIM;
        for (int r = 0; r < REPG; ++r) {
            float a = 0.f;
            for (int d = 0; d < HDIM; ++d) a += qsm[r][d] * kr[d];
            scm[r][t] = (pos <= s) ? a * scale : NEGV;
        }
    }
    __syncthreads();

    // softmax: wave w handles row w (4 waves of 32 lanes)
    {
        int wave = t >> 5, lane = t & 31;
        float vals[4];
        float mloc = -3.0e38f;
        for (int i = 0; i < 4; ++i) {
            vals[i] = scm[wave][lane + 32 * i];
            mloc = fmaxf(mloc, vals[i]);
        }
        for (int m = 1; m <= 16; m <<= 1)
            mloc = fmaxf(mloc, __shfl_xor(mloc, m, 32));
        float ssum = 0.f;
        for (int i = 0; i < 4; ++i) { vals[i] = expf(vals[i] - mloc); ssum += vals[i]; }
        for (int m = 1; m <= 16; m <<= 1)
            ssum += __shfl_xor(ssum, m, 32);
        float inv = 1.0f / ssum;
        for (int i = 0; i < 4; ++i) scm[wave][lane + 32 * i] = vals[i] * inv;
    }
    __syncthreads();

    // output: thread t handles dim d = t for all 4 rep heads
    {
        float o0 = 0.f, o1 = 0.f, o2 = 0.f, o3 = 0.f;
        for (int key = 0; key < TOPN * BLKC; ++key) {
            int pos = selb[key / BLKC] * BLKC + (key % BLKC);
            float vv = v[(pos * NHK + hk) * HDIM + t];
            o0 += scm[0][key] * vv;
            o1 += scm[1][key] * vv;
            o2 += scm[2][key] * vv;
            o3 += scm[3][key] * vv;
        }
        float os[4] = {o0, o1, o2, o3};
        for (int r = 0; r < REPG; ++r) {
            int hq = hk * REPG + r;
            out[(s * NHQ + hq) * HDIM + t] += gs[s * NHQ + hq] * os[r];
        }
    }
}

// --------------------------------------------------------------------------
extern "C" void kernel_launch(void* const* d_in, const int* in_sizes, int n_in,
                              void* d_out, int out_size, void* d_ws, size_t ws_size,
                              hipStream_t stream) {
    const float* q  = (const float*)d_in[0];
    const float* k  = (const float*)d_in[1];
    const float* v  = (const float*)d_in[2];
    const float* gw = (const float*)d_in[3];
    const float* gc = (const float*)d_in[4];
    const float* gs = (const float*)d_in[5];
    float* out = (float*)d_out;

    float* ws = (float*)d_ws;
    float* kc = ws;                                  // NB*NHK*HD = 16384 floats
    float* vc = ws + NBC * NHK * HDIM;               // 16384 floats
    float* pc = vc + NBC * NHK * HDIM;               // NHQ*S*NB = 1,048,576 floats
    int*  sel = (int*)(pc + (size_t)NHQ * SS * NBC); // S*NHK*TOPN ints

    nsa_cmpkv_kernel<<<(NBC * NHK * HDIM + 255) / 256, 256, 0, stream>>>(k, v, kc, vc);
    nsa_win_kernel<<<(SS / 16) * NHQ, 32, 0, stream>>>(q, k, v, gw, out);
    nsa_cmp_kernel<<<(SS / 16) * NHQ, 32, 0, stream>>>(q, kc, vc, gc, pc, out);
    nsa_sel_kernel<<<(SS * NHK + 255) / 256, 256, 0, stream>>>(pc, sel);
    nsa_slt_kernel<<<SS * NHK, 128, 0, stream>>>(q, k, v, gs, sel, out);
}